// HGATSelfAttention_40999757807672
// MI455X (gfx1250) — compile-verified
//
#include <hip/hip_runtime.h>

// ---- problem constants (from reference) ----
#define T_DIM  192
#define B_DIM  2
#define E_DIM  512
#define H_DIM  2
#define HD_DIM 256
#define BH_DIM 4

typedef __attribute__((ext_vector_type(2))) float v2f;
typedef __attribute__((ext_vector_type(8))) float v8f;
typedef __attribute__((ext_vector_type(4))) unsigned int u32x4;
typedef __attribute__((ext_vector_type(8))) int i32x8;
typedef __attribute__((ext_vector_type(4))) int i32x4;

// ---------------------------------------------------------------------------
// f32 WMMA GEMM, TDM-staged B panel.
// Block = 128 threads (4 waves). Each block computes a 64(M) x 16(N) tile;
// wave w owns rows [m0+16w, m0+16w+16). The K x 16 B panel is DMA'd into LDS
// once per block by the Tensor Data Mover (wave 0 issues the descriptor),
// then all 4 waves stream B fragments from LDS while A comes from global.
//
// WMMA fragment layouts (V_WMMA_F32_16X16X4_F32):
//   A (16x4):  lanes 0-15 -> M=lane, v0=K+0,v1=K+1 ; lanes16-31 -> v0=K+2,v1=K+3
//   B (4x16):  lanes 0-15 -> N=lane, v0=K+0,v1=K+1 ; lanes16-31 -> v0=K+2,v1=K+3
//   C/D(16x16):lanes 0-15 -> N=lane, VGPRv=M=v ; lanes16-31 -> M=v+8
// Requires N%16==0, K%4==0, M%64==0 for the grids used here.
// ---------------------------------------------------------------------------
__global__ __launch_bounds__(128)
void gemm_f32_wmma_tdm(const float* __restrict__ A, int lda,
                       const float* __restrict__ Bm, int ldb,
                       const float* __restrict__ bias,
                       float* __restrict__ C, int ldc,
                       int K, int Ntot)
{
    __shared__ __align__(16) float bpanel[512 * 16];   // K_max x 16 = 32 KB

    const int lane = threadIdx.x & 31;
    const int wave = threadIdx.x >> 5;
    const int half = lane >> 4;
    const int l16  = lane & 15;
    const int n0 = blockIdx.x << 4;
    const int m0 = (blockIdx.y << 6) + (wave << 4);

    if (wave == 0) {
        // ---- Tensor DMA descriptor (D#): 2-D tile, K rows x 16 cols of f32 ----
        const unsigned long long gaddr =
            (unsigned long long)(uintptr_t)(Bm + n0);          // tile start (bytes)
        const unsigned lds_off = (unsigned)(uintptr_t)(void*)bpanel; // LDS byte offset
        const unsigned td0 = (unsigned)(Ntot - n0);            // tensor dim0 (cols left)
        const unsigned td1 = (unsigned)K;                      // tensor dim1 (rows)

        u32x4 g0;
        g0[0] = 1u;                                            // count=1, user descriptor
        g0[1] = lds_off;                                       // lds_addr
        g0[2] = (unsigned)(gaddr & 0xffffffffu);               // global_addr[31:0]
        g0[3] = (unsigned)((gaddr >> 32) & 0x1ffffffu)         // global_addr[56:32]
              | (2u << 30);                                    // type=2 (image)

        i32x8 g1;
        g1[0] = (int)(2u << 16);                               // data_size=2 (4 bytes)
        g1[1] = (int)((td0 & 0xffffu) << 16);                  // tensor_dim0[15:0]
        g1[2] = (int)(((td0 >> 16) & 0xffffu)                  // tensor_dim0[31:16]
              | ((td1 & 0xffffu) << 16));                      // tensor_dim1[15:0]
        g1[3] = (int)(((td1 >> 16) & 0xffffu)                  // tensor_dim1[31:16]
              | (16u << 16));                                  // tile_dim0 = 16
        g1[4] = (int)(td1 & 0xffffu);                          // tile_dim1 = K, tile_dim2=0
        g1[5] = (int)(unsigned)ldb;                            // tensor_dim0_stride[31:0]
        g1[6] = 0;                                             // stride hi / dim1_stride lo
        g1[7] = 0;

        const i32x4 gz4 = {0, 0, 0, 0};                        // groups 2/3 unused (2-D)
        const i32x8 gz8 = {0, 0, 0, 0, 0, 0, 0, 0};
        __builtin_amdgcn_tensor_load_to_lds(g0, g1, gz4, gz4, gz8, 0);
        __builtin_amdgcn_s_wait_tensorcnt(0);
    }
    __syncthreads();                                           // B panel visible to block

    v8f acc;
    const float bv = bias ? bias[n0 + l16] : 0.0f;
#pragma unroll
    for (int i = 0; i < 8; ++i) acc[i] = bv;

    const float* Arow = A + (m0 + l16) * lda + half * 2;       // contiguous K pair
    const float* Bl   = bpanel + (half * 2) * 16 + l16;        // LDS B fragment base

#pragma unroll 8
    for (int k = 0; k < K; k += 4) {
        v2f a = *(const v2f*)(Arow + k);
        v2f b;
        b.x = Bl[k * 16];
        b.y = Bl[(k + 1) * 16];
        acc = __builtin_amdgcn_wmma_f32_16x16x4_f32(
                  /*neg_a=*/false, a, /*neg_b=*/false, b,
                  /*c_mod=*/(short)0, acc, /*reuse_a=*/false, /*reuse_b=*/false);
    }

    float* Crow = C + (m0 + half * 8) * ldc + n0 + l16;
#pragma unroll
    for (int v = 0; v < 8; ++v) Crow[v * ldc] = acc[v];
}

// ---------------------------------------------------------------------------
// Tiny per-row dot products against the three 256-slices of Wa[768,1].
// One wave per dot; 3*T*BH = 2304 dots of length 256.
//   which=0: P1[t*4+bh] = <(h? e : q)[t,b,:], Wa[0:256]>
//   which=1: P2[bh*T+t] = <(h? e : k)[t,b,:], Wa[256:512]>
//   which=2: Ee[t*4+bh] = <entity[t,b,h*256:...], Wa[512:768]>
// ---------------------------------------------------------------------------
__global__ __launch_bounds__(256)
void dots_kernel(const float* __restrict__ q, const float* __restrict__ k,
                 const float* __restrict__ e, const float* __restrict__ entity,
                 const float* __restrict__ Wa,
                 float* __restrict__ P1, float* __restrict__ P2,
                 float* __restrict__ Ee)
{
    const int wave = (blockIdx.x * blockDim.x + threadIdx.x) >> 5;
    const int lane = threadIdx.x & 31;
    if (wave >= 3 * T_DIM * BH_DIM) return;
    const int which = wave / (T_DIM * BH_DIM);
    const int idx   = wave % (T_DIM * BH_DIM);
    const int t = idx >> 2, bh = idx & 3, b = bh >> 1, h = bh & 1;

    const float* src;
    const float* w;
    if (which == 0)      { src = (h ? e : q) + (t * B_DIM + b) * HD_DIM; w = Wa; }
    else if (which == 1) { src = (h ? e : k) + (t * B_DIM + b) * HD_DIM; w = Wa + HD_DIM; }
    else                 { src = entity + (t * B_DIM + b) * E_DIM + h * HD_DIM; w = Wa + 2 * HD_DIM; }

    float s = 0.0f;
#pragma unroll
    for (int d = lane; d < HD_DIM; d += 32) s += src[d] * w[d];
#pragma unroll
    for (int off = 16; off; off >>= 1) s += __shfl_xor(s, off, 32);

    if (lane == 0) {
        if (which == 0)      P1[t * 4 + bh] = s;
        else if (which == 1) P2[bh * T_DIM + t] = s;
        else                 Ee[t * 4 + bh] = s;
    }
}

// ---------------------------------------------------------------------------
// score + leaky_relu + key-padding mask + row softmax.
// One block (192 threads) per (g,p) row; r = threadIdx.x.
// Index algebra from the raw .view reshapes:
//   i = 48g + p/4 ; j = 48(p%4) + r/4 ; bh = r%4 ; ke-term uses (p%4, r).
// ---------------------------------------------------------------------------
__global__ __launch_bounds__(192)
void score_softmax(const float* __restrict__ P1, const float* __restrict__ P2,
                   const float* __restrict__ Ee, const float* __restrict__ ba,
                   const int* __restrict__ mask, float* __restrict__ probs)
{
    const int row = blockIdx.x;            // 0..767  (= g*T + p)
    const int g = row / T_DIM, p = row % T_DIM;
    const int r = threadIdx.x;
    const int i  = 48 * g + (p >> 2);
    const int bh = r & 3;
    const int j  = 48 * (p & 3) + (r >> 2);

    float s = P1[i * 4 + bh] + P2[(p & 3) * T_DIM + r]
            + Ee[i * 4 + bh] - Ee[j * 4 + bh] + ba[0];
    s = (s > 0.0f) ? s : 0.01f * s;                       // leaky_relu, slope .01
    if (mask[(g >> 1) * T_DIM + r]) s = -__builtin_inff();

    __shared__ float red[6];
    const int lane = r & 31, warp = r >> 5;

    float m = s;
#pragma unroll
    for (int off = 16; off; off >>= 1) m = fmaxf(m, __shfl_xor(m, off, 32));
    if (lane == 0) red[warp] = m;
    __syncthreads();
    m = red[0];
#pragma unroll
    for (int w = 1; w < 6; ++w) m = fmaxf(m, red[w]);

    const float ex = __expf(s - m);
    __syncthreads();
    float sum = ex;
#pragma unroll
    for (int off = 16; off; off >>= 1) sum += __shfl_xor(sum, off, 32);
    if (lane == 0) red[warp] = sum;
    __syncthreads();
    sum = red[0] + red[1] + red[2] + red[3] + red[4] + red[5];

    probs[row * T_DIM + r] = ex / sum;
}

// ---------------------------------------------------------------------------
// Host-side orchestration.
// ---------------------------------------------------------------------------
extern "C" void kernel_launch(void* const* d_in, const int* in_sizes, int n_in,
                              void* d_out, int out_size, void* d_ws, size_t ws_size,
                              hipStream_t stream)
{
    (void)in_sizes; (void)n_in; (void)out_size; (void)ws_size;

    const float* query  = (const float*)d_in[0];
    const float* entity = (const float*)d_in[1];
    // d_in[2..4]: key / ex_relation / ex_entity — dead code in the reference.
    const int*   kpm    = (const int*)d_in[5];
    const float* Wq = (const float*)d_in[6];   const float* bq = (const float*)d_in[7];
    const float* Wk = (const float*)d_in[8];   const float* bk = (const float*)d_in[9];
    const float* Wv = (const float*)d_in[10];  const float* bv = (const float*)d_in[11];
    const float* We = (const float*)d_in[12];  const float* be = (const float*)d_in[13];
    // d_in[14..17]: Wxe/bxe/Wxr/bxr — dead code.
    const float* Wa = (const float*)d_in[18];  const float* ba = (const float*)d_in[19];
    const float* Wo = (const float*)d_in[20];  const float* bo = (const float*)d_in[21];
    float* out = (float*)d_out;

    // workspace carve-up (floats)
    float* ws    = (float*)d_ws;
    float* q_ws  = ws;                                   // 384*256
    float* k_ws  = q_ws + T_DIM * B_DIM * HD_DIM;
    float* v_ws  = k_ws + T_DIM * B_DIM * HD_DIM;
    float* e_ws  = v_ws + T_DIM * B_DIM * HD_DIM;
    float* P1    = e_ws + T_DIM * B_DIM * HD_DIM;        // 768
    float* P2    = P1 + T_DIM * BH_DIM;                  // 768
    float* Ee    = P2 + T_DIM * BH_DIM;                  // 768
    float* probs = Ee + T_DIM * BH_DIM;                  // 4*192*192
    float* attn  = probs + BH_DIM * T_DIM * T_DIM;       // 384*512

    const dim3 blk(128);                                 // 4 waves / block

    // 1) projections: [384,512] @ [512,256] + bias   (M=384 -> 6 row-blocks of 64)
    const dim3 gproj(HD_DIM / 16, (T_DIM * B_DIM) / 64);           // 16 x 6
    gemm_f32_wmma_tdm<<<gproj, blk, 0, stream>>>(query,  E_DIM, Wq, HD_DIM, bq, q_ws, HD_DIM, E_DIM, HD_DIM);
    gemm_f32_wmma_tdm<<<gproj, blk, 0, stream>>>(query,  E_DIM, Wk, HD_DIM, bk, k_ws, HD_DIM, E_DIM, HD_DIM);
    gemm_f32_wmma_tdm<<<gproj, blk, 0, stream>>>(query,  E_DIM, Wv, HD_DIM, bv, v_ws, HD_DIM, E_DIM, HD_DIM);
    gemm_f32_wmma_tdm<<<gproj, blk, 0, stream>>>(entity, E_DIM, We, HD_DIM, be, e_ws, HD_DIM, E_DIM, HD_DIM);

    // 2) rank-1 score components (Wa is [768,1] -> three 256-dot families)
    dots_kernel<<<288, 256, 0, stream>>>(q_ws, k_ws, e_ws, entity, Wa, P1, P2, Ee);

    // 3) scores + mask + softmax -> probs [BH,T,T]
    score_softmax<<<BH_DIM * T_DIM, T_DIM, 0, stream>>>(P1, P2, Ee, ba, kpm, probs);

    // 4) out[g] = probs[g] @ ve_s[g]  (head0 -> v, head1 -> e), written directly
    //    into attn laid out as [T, B, E] (t-major rows, head h in columns h*HD..)
    for (int g = 0; g < BH_DIM; ++g) {
        const int b = g >> 1, h = g & 1;
        gemm_f32_wmma_tdm<<<dim3(HD_DIM / 16, T_DIM / 64), blk, 0, stream>>>(
            probs + g * T_DIM * T_DIM, T_DIM,
            (h ? e_ws : v_ws) + b * HD_DIM, B_DIM * HD_DIM,   // gather via base+ld
            nullptr,
            attn + b * E_DIM + h * HD_DIM, B_DIM * E_DIM,
            T_DIM, HD_DIM);
    }

    // 5) final: out = attn[384,512] @ Wo[512,512] + bo
    gemm_f32_wmma_tdm<<<dim3(E_DIM / 16, (T_DIM * B_DIM) / 64), blk, 0, stream>>>(
        attn, E_DIM, Wo, E_DIM, bo, out, E_DIM, E_DIM, E_DIM);
}